// Sampler_30760555774542
// MI455X (gfx1250) — compile-verified
//
#include <hip/hip_runtime.h>
#include <hip/hip_bf16.h>
#include <stdint.h>

#define THREADS 256
#define WAVES   8                    // wave32 on gfx1250
#define DEPTH   12                   // async-to-LDS pipeline depth (ring buffers / wave)
#define CHUNK   (THREADS * 4)        // 1024 floats per block iteration
#define SPLIT   8                    // workgroups per row for the HBM pass
#define CAP     512                  // candidate buffer (need top-64; few hundred typical)
#define NBINS   1024
#define BIN_SCALE 64.0f              // bins per unit below row max; range = 16.0

static_assert(DEPTH == 12, "WAIT_ASYNC immediates below assume DEPTH==12");

// ---- CDNA5 async-to-LDS (ASYNCcnt) helpers -------------------------------
#define WAIT_ASYNC(n) asm volatile("s_wait_asynccnt " #n ::: "memory")

__device__ __forceinline__ void async_b128(uint32_t lds_addr, uint64_t gaddr) {
    // GV mode: per-lane 64-bit global address, per-lane LDS dest address.
    asm volatile("global_load_async_to_lds_b128 %0, %1, off"
                 :: "v"(lds_addr), "v"(gaddr) : "memory");
}

// monotonic float -> sortable uint mapping
__device__ __forceinline__ uint32_t f2key(float f) {
    uint32_t u = __float_as_uint(f);
    return (u & 0x80000000u) ? ~u : (u | 0x80000000u);
}
__device__ __forceinline__ float key2f(uint32_t k) {
    uint32_t u = (k & 0x80000000u) ? (k & 0x7fffffffu) : ~k;
    return __uint_as_float(u);
}

__device__ __forceinline__ unsigned long long splitmix64(unsigned long long z) {
    z += 0x9E3779B97F4A7C15ull;
    z = (z ^ (z >> 30)) * 0xBF58476D1CE4E5B9ull;
    z = (z ^ (z >> 27)) * 0x94D049BB133111EBull;
    return z ^ (z >> 31);
}

// Deeply pipelined streaming over [base, base+n) through LDS staging buffers.
// Each wave keeps up to DEPTH async b128 loads (512B tiles) outstanding.
// fnv(float4, idx0) consumes 4 elements at idx0..idx0+3; fns(float, idx) scalar tail.
template <typename FV, typename FS>
__device__ __forceinline__ void stream_async(const float* __restrict__ base, int n,
                                             float4* __restrict__ stage,
                                             int wave, int lane, int tid,
                                             FV fnv, FS fns)
{
    int nfull = n / CHUNK;
    if (nfull < DEPTH) nfull = 0;           // pipeline needs >= DEPTH chunks
    if (nfull > 0) {
        float4* ws_ = stage + wave * (DEPTH * 32);
        #pragma unroll
        for (int k = 0; k < DEPTH; ++k) {
            async_b128((uint32_t)(uintptr_t)&ws_[k * 32 + lane],
                       (uint64_t)(uintptr_t)(base + (size_t)k * CHUNK + wave * 128 + lane * 4));
        }
        int slot = 0;
        const int main_iters = nfull - DEPTH;
        for (int i = 0; i < main_iters; ++i) {
            WAIT_ASYNC(11);                 // DEPTH-1 outstanding => oldest tile landed
            float4 v = ws_[slot * 32 + lane];
            fnv(v, i * CHUNK + wave * 128 + lane * 4);
            async_b128((uint32_t)(uintptr_t)&ws_[slot * 32 + lane],
                       (uint64_t)(uintptr_t)(base + (size_t)(i + DEPTH) * CHUNK + wave * 128 + lane * 4));
            slot = (slot + 1 == DEPTH) ? 0 : slot + 1;
        }
        WAIT_ASYNC(0);                      // drain: everything issued has landed
        for (int i = main_iters; i < nfull; ++i) {
            float4 v = ws_[slot * 32 + lane];
            fnv(v, i * CHUNK + wave * 128 + lane * 4);
            slot = (slot + 1 == DEPTH) ? 0 : slot + 1;
        }
    }
    for (int idx = nfull * CHUNK + tid; idx < n; idx += THREADS)
        fns(base[idx], idx);
}

// =========================================================================
// Pass 1 (HBM-bound): SPLIT workgroups per row, partial online max / sumexp.
// =========================================================================
__global__ __launch_bounds__(THREADS)
void sampler_pass1(const float* __restrict__ logits,
                   const float* __restrict__ temps,
                   float* __restrict__ ws_m, float* __restrict__ ws_s, int V)
{
    __shared__ float4 stage[WAVES * DEPTH * 32];    // 48 KB staging
    __shared__ float red_m[THREADS], red_s[THREADS];

    const int wg   = blockIdx.x;
    const int b    = wg / SPLIT;
    const int j    = wg % SPLIT;
    const int tid  = threadIdx.x;
    const int wave = tid >> 5;
    const int lane = tid & 31;
    const float invT = 1.0f / temps[b];
    const int seg = V / SPLIT;
    const int off = j * seg;
    const int len = (j == SPLIT - 1) ? (V - off) : seg;
    const float* base = logits + (size_t)b * (size_t)V + off;

    float m = -3.4e38f, s = 0.0f;
    auto upd = [&](float x) {
        float sc = x * invT;
        if (sc > m) { s = s * __expf(m - sc) + 1.0f; m = sc; }
        else        { s += __expf(sc - m); }
    };
    stream_async(base, len, stage, wave, lane, tid,
        [&](float4 v, int) { upd(v.x); upd(v.y); upd(v.z); upd(v.w); },
        [&](float x, int)  { upd(x); });

    red_m[tid] = m; red_s[tid] = s;
    __syncthreads();
    for (int o = THREADS / 2; o > 0; o >>= 1) {
        if (tid < o) {
            float m1 = red_m[tid], s1 = red_s[tid];
            float m2 = red_m[tid + o], s2 = red_s[tid + o];
            float mm = fmaxf(m1, m2);
            red_s[tid] = s1 * __expf(m1 - mm) + s2 * __expf(m2 - mm);
            red_m[tid] = mm;
        }
        __syncthreads();
    }
    if (tid == 0) { ws_m[wg] = red_m[0]; ws_s[wg] = red_s[0]; }
}

// =========================================================================
// Pass 2 (L2-resident): combine partials, histogram -> top-64 threshold,
// extract candidates, sort, apply top-k/top-p keep rule, sample.
// =========================================================================
__global__ __launch_bounds__(THREADS)
void sampler_pass2(const float* __restrict__ logits,
                   const float* __restrict__ temps,
                   const float* __restrict__ toppv,
                   const int*   __restrict__ topkv,
                   const float* __restrict__ ws_m,
                   const float* __restrict__ ws_s,
                   int* __restrict__ out, int V)
{
    __shared__ float4 stage[WAVES * DEPTH * 32];    // 48 KB staging
    __shared__ unsigned hist[NBINS];                // 4 KB
    __shared__ unsigned long long cand[CAP];        // 4 KB
    __shared__ unsigned s_cnt;
    __shared__ float s_mv, s_sv, s_thr;
    __shared__ float kp[64];
    __shared__ int   ki[64];

    const int b    = blockIdx.x;
    const int tid  = threadIdx.x;
    const int wave = tid >> 5;
    const int lane = tid & 31;
    const float invT = 1.0f / temps[b];
    const float* row = logits + (size_t)b * (size_t)V;

    if (tid == 0) {
        float mm = ws_m[b * SPLIT], ss = ws_s[b * SPLIT];
        for (int j = 1; j < SPLIT; ++j) {
            float m2 = ws_m[b * SPLIT + j], s2 = ws_s[b * SPLIT + j];
            float mx = fmaxf(mm, m2);
            ss = ss * __expf(mm - mx) + s2 * __expf(m2 - mx);
            mm = mx;
        }
        s_mv = mm; s_sv = ss; s_cnt = 0u;
    }
    for (int i = tid; i < NBINS; i += THREADS) hist[i] = 0u;
    __syncthreads();
    const float rowm = s_mv;
    const float rows = s_sv;

    // ---- sweep A: relative histogram below the row max --------------------
    auto histo1 = [&](float x) {
        float sc = x * invT;
        float d  = (rowm - sc) * BIN_SCALE;
        if (d < (float)NBINS) {
            int bin = (int)d; if (bin < 0) bin = 0;
            atomicAdd(&hist[bin], 1u);
        }
    };
    stream_async(row, V, stage, wave, lane, tid,
        [&](float4 v, int) { histo1(v.x); histo1(v.y); histo1(v.z); histo1(v.w); },
        [&](float x, int)  { histo1(x); });
    __syncthreads();

    if (tid == 0) {
        unsigned cum = 0; int bstar = NBINS - 1;
        for (int i = 0; i < NBINS; ++i) {
            cum += hist[i];
            if (cum >= 64u) { bstar = i; break; }
        }
        s_thr = rowm - (float)(bstar + 1) * (1.0f / BIN_SCALE);
    }
    __syncthreads();
    const float thr = s_thr;

    // ---- sweep B: extract candidates >= threshold -------------------------
    auto extract1 = [&](float x, int idx) {
        float sc = x * invT;
        if (sc >= thr) {
            unsigned pos = atomicAdd(&s_cnt, 1u);
            if (pos < CAP)
                cand[pos] = ((unsigned long long)f2key(sc) << 32) | (unsigned)idx;
        }
    };
    stream_async(row, V, stage, wave, lane, tid,
        [&](float4 v, int i0) {
            extract1(v.x, i0); extract1(v.y, i0 + 1);
            extract1(v.z, i0 + 2); extract1(v.w, i0 + 3);
        },
        [&](float x, int idx) { extract1(x, idx); });
    __syncthreads();

    unsigned ccount = s_cnt; if (ccount > CAP) ccount = CAP;
    for (unsigned i = ccount + tid; i < CAP; i += THREADS) cand[i] = 0ull;
    __syncthreads();

    // bitonic sort, descending by packed (sortable_key, index)
    for (unsigned k = 2; k <= CAP; k <<= 1) {
        for (unsigned j = k >> 1; j > 0; j >>= 1) {
            for (unsigned t = tid; t < CAP; t += THREADS) {
                unsigned x = t ^ j;
                if (x > t) {
                    unsigned long long a = cand[t], c2 = cand[x];
                    bool desc = ((t & k) == 0);
                    if (desc ? (a < c2) : (a > c2)) { cand[t] = c2; cand[x] = a; }
                }
            }
            __syncthreads();
        }
    }

    if (tid == 0) {
        const float tp = toppv[b];
        const int   tk = topkv[b];
        int K = (int)ccount; if (K > 64) K = 64;
        float cum = 0.0f; int nk = 0;
        for (int r = 0; r < K; ++r) {
            unsigned long long e = cand[r];
            float sc = key2f((uint32_t)(e >> 32));
            float p  = __expf(sc - rowm) / rows;
            // reference: keep = (rank < top_k) && (cum_before < top_p); rank 0 forced
            bool keep = (r == 0) || ((r < tk) && (cum < tp));
            cum += p;
            if (keep) { kp[nk] = p; ki[nk] = (int)(e & 0xffffffffull); ++nk; }
        }
        float Z = 0.0f;
        for (int i = 0; i < nk; ++i) Z += kp[i];
        unsigned long long h =
            splitmix64(0x2545F4914F6CDD1Dull ^ (0x9E3779B97F4A7C15ull * (unsigned long long)(b + 1)) ^ 42ull);
        float u = (float)(h >> 40) * (1.0f / 16777216.0f);
        float rsel = u * Z, acc = 0.0f;
        int tok = (nk > 0) ? ki[0] : 0;
        for (int i = 0; i < nk; ++i) { acc += kp[i]; if (acc >= rsel) { tok = ki[i]; break; } }
        out[b] = tok;
    }
}

extern "C" void kernel_launch(void* const* d_in, const int* in_sizes, int n_in,
                              void* d_out, int out_size, void* d_ws, size_t ws_size,
                              hipStream_t stream) {
    const float* logits = (const float*)d_in[0];
    const float* temps  = (const float*)d_in[1];
    const float* topp   = (const float*)d_in[2];
    const int*   topk   = (const int*)d_in[3];
    const int B = in_sizes[1];
    const int V = in_sizes[0] / B;
    int* out = (int*)d_out;

    float* ws_m = (float*)d_ws;                 // B*SPLIT partial maxes
    float* ws_s = ws_m + (size_t)B * SPLIT;     // B*SPLIT partial sumexp

    sampler_pass1<<<B * SPLIT, THREADS, 0, stream>>>(logits, temps, ws_m, ws_s, V);
    sampler_pass2<<<B, THREADS, 0, stream>>>(logits, temps, topp, topk, ws_m, ws_s, out, V);

    (void)n_in; (void)out_size; (void)ws_size;
}